// TimeGSMKernel_45457933861304
// MI455X (gfx1250) — compile-verified
//
#include <hip/hip_runtime.h>

typedef float v2f __attribute__((ext_vector_type(2)));
typedef float v8f __attribute__((ext_vector_type(8)));

#define TGSM_B 4
#define TGSM_L 1024
#define FEAT_STRIDE 16   // 13 used floats per point, padded to 16 for b128 I/O
#define IT 32            // i-rows per block in pair stage
#define JT 256           // j-cols per block in pair stage

__device__ __forceinline__ float relu_(float x) { return fmaxf(x, 0.0f); }

// Second dense layer (4x4) of one FFN for 16 points, via V_WMMA_F32_16X16X4_F32.
// A (16x4 f32): lane = M (both halves), VGPR0/1 = K{0,1} (lanes 0-15) / K{2,3} (lanes 16-31).
// B (4x16 f32): lane = N, same K split across halves. Bias enters through C (broadcast per column).
// D (16x16 f32): lanes 0-15 VGPR r -> row r, lanes 16-31 VGPR r -> row 8+r; scattered to LDS.
__device__ __forceinline__ void wmma_ffn2(const float* h, const float* __restrict__ W2,
                                          const float* __restrict__ b2,
                                          int m, int hi, float* ldsRow) {
  const int k0 = hi * 2;
  v2f a;
  a[0] = hi ? h[2] : h[0];
  a[1] = hi ? h[3] : h[1];
  v2f bm;
  bm[0] = 0.0f;
  bm[1] = 0.0f;
  float cv = 0.0f;
  if (m < 4) {                       // only columns 0..3 are real (W2 is 4x4)
    bm[0] = W2[k0 * 4 + m];
    bm[1] = W2[(k0 + 1) * 4 + m];
    cv = b2[m];                      // bias rides in as the C matrix
  }
  v8f c;
#pragma unroll
  for (int r = 0; r < 8; ++r) c[r] = cv;
  v8f d = __builtin_amdgcn_wmma_f32_16x16x4_f32(false, a, false, bm,
                                                (short)0, c, false, false);
#pragma unroll
  for (int r = 0; r < 8; ++r)
    ldsRow[(hi * 8 + r) * 16 + m] = d[r];
}

// Stage 1: per-point FFNs (layer2 via WMMA) -> factorized pair features.
// feat[p*16 + {0..3}]  = s_d^2
// feat[p*16 + {4..7}]  = sqrt(s_d)*be_d*cos(2*pi*pv_d*t)
// feat[p*16 + {8..11}] = sqrt(s_d)*be_d*sin(2*pi*pv_d*t)
// feat[p*16 + 12]      = t
__global__ __launch_bounds__(256) void tgsm_features_kernel(
    const float* __restrict__ t,
    const float* __restrict__ Wp1, const float* __restrict__ bp1,
    const float* __restrict__ Wp2, const float* __restrict__ bp2,
    const float* __restrict__ Ws1, const float* __restrict__ bs1,
    const float* __restrict__ Ws2, const float* __restrict__ bs2,
    const float* __restrict__ Wb1, const float* __restrict__ bb1,
    const float* __restrict__ Wb2, const float* __restrict__ bb2,
    float* __restrict__ feat) {
  __shared__ float ldsD[3][8][256];

  const int tid  = threadIdx.x;
  const int wave = tid >> 5;        // wave32
  const int lane = tid & 31;
  const int m    = lane & 15;       // point index within the wave's 16 points
  const int hi   = lane >> 4;       // half-wave selects K pair
  const int gp   = (blockIdx.x * 8 + wave) * 16 + m;   // 0..4095 == b*L + l
  const float tv = t[gp];

  // Layer 1 (1 -> 4) for all three FFNs: uniform weights -> s_loads.
  float hp[4], hs[4], hb[4];
#pragma unroll
  for (int k = 0; k < 4; ++k) {
    hp[k] = relu_(fmaf(tv, Wp1[k], bp1[k]));
    hs[k] = relu_(fmaf(tv, Ws1[k], bs1[k]));
    hb[k] = relu_(fmaf(tv, Wb1[k], bb1[k]));
  }

  // Layer 2 (4 -> 4) on the matrix core.
  wmma_ffn2(hp, Wp2, bp2, m, hi, &ldsD[0][wave][0]);
  wmma_ffn2(hs, Ws2, bs2, m, hi, &ldsD[1][wave][0]);
  wmma_ffn2(hb, Wb2, bb2, m, hi, &ldsD[2][wave][0]);
  __syncthreads();

  if (lane < 16) {
    const float4 dp = *(const float4*)&ldsD[0][wave][m * 16];  // row m, cols 0..3
    const float4 ds = *(const float4*)&ldsD[1][wave][m * 16];
    const float4 db = *(const float4*)&ldsD[2][wave][m * 16];
    const float pv[4] = {relu_(dp.x), relu_(dp.y), relu_(dp.z), relu_(dp.w)};
    const float sg[4] = {relu_(ds.x) + 1e-6f, relu_(ds.y) + 1e-6f,
                         relu_(ds.z) + 1e-6f, relu_(ds.w) + 1e-6f};
    const float be[4] = {relu_(db.x), relu_(db.y), relu_(db.z), relu_(db.w)};

    float o[16];
#pragma unroll
    for (int d = 0; d < 4; ++d) {
      const float q  = pv[d] * tv;                  // qp/(2*pi): v_sin/v_cos take revolutions
      const float cq = __builtin_amdgcn_cosf(q);    // cos(2*pi*pv*t)
      const float sq = __builtin_amdgcn_sinf(q);
      const float rt = __builtin_amdgcn_sqrtf(sg[d]);
      const float w  = rt * be[d];
      o[d]     = sg[d] * sg[d];
      o[4 + d] = w * cq;
      o[8 + d] = w * sq;
    }
    o[12] = tv; o[13] = 0.0f; o[14] = 0.0f; o[15] = 0.0f;

    float4* dst = (float4*)(feat + (size_t)gp * FEAT_STRIDE);
    dst[0] = make_float4(o[0],  o[1],  o[2],  o[3]);
    dst[1] = make_float4(o[4],  o[5],  o[6],  o[7]);
    dst[2] = make_float4(o[8],  o[9],  o[10], o[11]);
    dst[3] = make_float4(o[12], o[13], o[14], o[15]);
  }
}

// Stage 2: kernel[b,i,j] = sqrt(2) * sum_d (WcI*WcJ + WsI*WsJ) * rsqrt(sI2+sJ2)
//                                   * exp2(-log2(e)*(ti-tj)^2 * (1/(sI2+sJ2)))
// Per (pair,d): 1 v_rsq_f32 + 1 v_exp_f32 + ~6 mul/fma. No per-pair cos/sqrt/div.
__global__ __launch_bounds__(256) void tgsm_pair_kernel(const float* __restrict__ feat,
                                                        float* __restrict__ out) {
  __shared__ float Fi[IT * FEAT_STRIDE];

  const int tid   = threadIdx.x;
  const int b     = blockIdx.z;
  const int j     = blockIdx.x * JT + tid;
  const int ibase = blockIdx.y * IT;

  // This thread's j-features: 4 x global_load_b128, kept in VGPRs for all 32 i-rows.
  const float4* fj = (const float4*)(feat + ((size_t)(b * TGSM_L + j)) * FEAT_STRIDE);
  const float4 s2j = fj[0];
  const float4 wcj = fj[1];
  const float4 wsj = fj[2];
  const float  tj  = fj[3].x;

  // Stage the i-tile in LDS (broadcast reads in the hot loop).
  const float* fsrc = feat + ((size_t)(b * TGSM_L + ibase)) * FEAT_STRIDE;
  for (int idx = tid; idx < IT * FEAT_STRIDE; idx += 256) Fi[idx] = fsrc[idx];
  __syncthreads();

  float* orow = out + ((size_t)(b * TGSM_L + ibase)) * TGSM_L + j;

#define TGSM_PAIR_D(k, comp)                                              \
  {                                                                       \
    const float denom = fi[k] + s2j.comp;                                 \
    const float rs    = __builtin_amdgcn_rsqf(denom);                     \
    const float e     = __builtin_amdgcn_exp2f(ntd2 * (rs * rs));         \
    const float dot   = fmaf(fi[4 + k], wcj.comp, fi[8 + k] * wsj.comp);  \
    acc = fmaf(dot * rs, e, acc);                                         \
  }

#pragma unroll 4
  for (int ii = 0; ii < IT; ++ii) {
    const float* fi  = &Fi[ii * FEAT_STRIDE];
    const float ti   = fi[12];
    const float td   = ti - tj;
    const float ntd2 = -1.4426950408889634f * (td * td);  // -log2(e)*td^2
    float acc = 0.0f;
    TGSM_PAIR_D(0, x)
    TGSM_PAIR_D(1, y)
    TGSM_PAIR_D(2, z)
    TGSM_PAIR_D(3, w)
    orow[(size_t)ii * TGSM_L] = acc * 1.4142135623730951f;  // sqrt(2)
  }
#undef TGSM_PAIR_D
}

extern "C" void kernel_launch(void* const* d_in, const int* in_sizes, int n_in,
                              void* d_out, int out_size, void* d_ws, size_t ws_size,
                              hipStream_t stream) {
  const float* t   = (const float*)d_in[0];
  const float* Wp1 = (const float*)d_in[1];
  const float* bp1 = (const float*)d_in[2];
  const float* Wp2 = (const float*)d_in[3];
  const float* bp2 = (const float*)d_in[4];
  const float* Ws1 = (const float*)d_in[5];
  const float* bs1 = (const float*)d_in[6];
  const float* Ws2 = (const float*)d_in[7];
  const float* bs2 = (const float*)d_in[8];
  const float* Wb1 = (const float*)d_in[9];
  const float* bb1 = (const float*)d_in[10];
  const float* Wb2 = (const float*)d_in[11];
  const float* bb2 = (const float*)d_in[12];

  float* feat = (float*)d_ws;  // needs B*L*16*4 = 256 KB of scratch

  // Stage 1: 4096 points, 128 points/block (8 waves x 16 points).
  tgsm_features_kernel<<<dim3((TGSM_B * TGSM_L) / 128), 256, 0, stream>>>(
      t, Wp1, bp1, Wp2, bp2, Ws1, bs1, Ws2, bs2, Wb1, bb1, Wb2, bb2, feat);

  // Stage 2: 4 j-tiles x 32 i-tiles x 4 batches = 512 blocks.
  tgsm_pair_kernel<<<dim3(TGSM_L / JT, TGSM_L / IT, TGSM_B), 256, 0, stream>>>(
      feat, (float*)d_out);
}